// RecommenderModel_73452530696646
// MI455X (gfx1250) — compile-verified
//
#include <hip/hip_runtime.h>

// ---------------------------------------------------------------------------
// RecommenderModel for MI455X (gfx1250, wave32)
//  1) embed gather-sum          (bandwidth bound, L2-resident tables)
//  2) SpMM edge scatter-add     (f32 atomics; x & agg fit in 192MB L2)
//  3) out = agg @ W^T + b       via V_WMMA_F32_16X16X4_F32 with the shared
//     A-strip staged in LDS (conflict-free banking) and 2-strip B reuse.
// ---------------------------------------------------------------------------

typedef __attribute__((ext_vector_type(2))) float v2f;
typedef __attribute__((ext_vector_type(8))) float v8f;

#define N_NODES 100000
#define N_EDGES 600000
#define DIM     128

// LDS row pitch for the staged A strip: 132 floats so that consecutive
// matrix rows land 4 banks apart (132 mod 64 = 4) -> the 32 lanes of a
// ds_load_b64 A-fragment read touch all 64 banks exactly once.
#define LDS_PITCH 132
#define ROWS_PER_BLOCK 32     // two 16-row WMMA strips per block

// ---------------------------------------------------------------------------
// Zero the aggregation buffer (12.8M floats) with wide stores.
// ---------------------------------------------------------------------------
__global__ void zero_f32_kernel(float* __restrict__ p, int n4) {
    int i = blockIdx.x * blockDim.x + threadIdx.x;
    if (i < n4) {
        *(float4*)(p + (size_t)i * 4) = make_float4(0.f, 0.f, 0.f, 0.f);
    }
}

// ---------------------------------------------------------------------------
// x[n][d] = user_table[user[n]][d] + item_table[item[n]][d]
//         + behavior_table[behavior[n]][d]
// 32 lanes per node, float4 per lane (128 dims).
// ---------------------------------------------------------------------------
__global__ void embed_kernel(const int*   __restrict__ user,
                             const int*   __restrict__ item,
                             const int*   __restrict__ behavior,
                             const float* __restrict__ user_table,
                             const float* __restrict__ item_table,
                             const float* __restrict__ behavior_table,
                             float*       __restrict__ x) {
    int t    = blockIdx.x * blockDim.x + threadIdx.x;
    int node = t >> 5;              // 32 lanes per node
    int dg   = (t & 31) << 2;       // 4-dim group
    if (node >= N_NODES) return;

    int u = user[node];
    int i = item[node];
    int b = behavior[node];

    float4 a = *(const float4*)(user_table     + (size_t)u * DIM + dg);
    float4 c = *(const float4*)(item_table     + (size_t)i * DIM + dg);
    float4 d = *(const float4*)(behavior_table + (size_t)b * DIM + dg);

    float4 r = make_float4(a.x + c.x + d.x,
                           a.y + c.y + d.y,
                           a.z + c.z + d.z,
                           a.w + c.w + d.w);
    *(float4*)(x + (size_t)node * DIM + dg) = r;
}

// ---------------------------------------------------------------------------
// agg[rows[e]][d] += vals[e] * x[cols[e]][d]
// 32 lanes per edge, each lane: one float4 gather + 4 f32 global atomics.
// Both x and agg are L2-resident (51.2MB each vs 192MB L2), so the random
// gather/scatter stays on-chip.
// ---------------------------------------------------------------------------
__global__ void spmm_scatter_kernel(const int*   __restrict__ rows,
                                    const int*   __restrict__ cols,
                                    const float* __restrict__ vals,
                                    const float* __restrict__ x,
                                    float*       __restrict__ agg) {
    int t  = blockIdx.x * blockDim.x + threadIdx.x;
    int e  = t >> 5;
    int dg = (t & 31) << 2;
    if (e >= N_EDGES) return;

    int   r = rows[e];
    int   c = cols[e];
    float v = vals[e];

    float4 xv = *(const float4*)(x + (size_t)c * DIM + dg);
    float* dst = agg + (size_t)r * DIM + dg;
    atomicAdd(dst + 0, v * xv.x);
    atomicAdd(dst + 1, v * xv.y);
    atomicAdd(dst + 2, v * xv.z);
    atomicAdd(dst + 3, v * xv.w);
}

// ---------------------------------------------------------------------------
// out = agg @ W^T + b   via V_WMMA_F32_16X16X4_F32.
//
// Block = 256 threads = 8 waves, covering a 32-row x 128-col output panel:
//   * The 32x128 agg panel is staged ONCE into LDS (pitch 132 -> bank
//     conflict-free ds_load_b64 A-fragments), shared by all 8 waves
//     (kills the 8x redundant L2 reads of agg).
//   * Each wave owns one 16-col tile and TWO 16-row strips: every B
//     fragment (from W, L2-resident) feeds 2 WMMAs -> halves W traffic
//     and gives two independent accumulation chains to hide WMMA latency.
//
// Fragment layouts per CDNA5 ISA:
//   A (16x4 f32, 2 VGPR): lane L = row L%16; K {0,1} lanes 0-15,
//                         K {2,3} lanes 16-31 -> contiguous float2.
//   B (4x16 f32, 2 VGPR): lane L = col L%16, same K split;
//                         B[k][n] = W[n][k] -> contiguous float2 in W row.
//   C/D (8 VGPR): lane L col L%16; VGPR v row v + (L<16 ? 0 : 8).
// ---------------------------------------------------------------------------
__global__ void gemm_bias_wmma_kernel(const float* __restrict__ agg,
                                      const float* __restrict__ W,
                                      const float* __restrict__ bias,
                                      float*       __restrict__ out) {
    __shared__ float aLds[ROWS_PER_BLOCK * LDS_PITCH];   // 16.9 KB

    const int tid   = threadIdx.x;
    const int wave  = tid >> 5;                  // 0..7 -> column tile
    const int lane  = tid & 31;
    const int tileM = blockIdx.x * ROWS_PER_BLOCK;   // 3125 blocks exactly
    const int tileN = wave << 4;
    const int l16   = lane & 15;
    const int khalf = (lane < 16) ? 0 : 2;

    // ---- stage the 32x128 agg panel into LDS (coalesced float4) ----------
#pragma unroll
    for (int j = 0; j < 4; ++j) {
        int idx  = tid + j * 256;                // 0..1023 float4 slots
        int row  = idx >> 5;                     // 32 float4 per row
        int col4 = (idx & 31) << 2;
        float4 vdat = *(const float4*)(agg + (size_t)(tileM + row) * DIM + col4);
        *(float4*)(aLds + row * LDS_PITCH + col4) = vdat;
    }
    __syncthreads();

    const float* wrow  = W + (size_t)(tileN + l16) * DIM;       // B = W^T
    const float* aRow0 = aLds + (size_t)l16        * LDS_PITCH; // strip 0
    const float* aRow1 = aLds + (size_t)(16 + l16) * LDS_PITCH; // strip 1

    v8f c0 = {};
    v8f c1 = {};
#pragma unroll
    for (int kk = 0; kk < DIM; kk += 4) {
        v2f b  = *(const v2f*)(wrow  + kk + khalf);   // global (L2)
        v2f a0 = *(const v2f*)(aRow0 + kk + khalf);   // LDS, conflict-free
        v2f a1 = *(const v2f*)(aRow1 + kk + khalf);
        // (neg_a, A, neg_b, B, c_mod, C, reuse_a, reuse_b)
        c0 = __builtin_amdgcn_wmma_f32_16x16x4_f32(
                 false, a0, false, b, (short)0, c0, false, false);
        c1 = __builtin_amdgcn_wmma_f32_16x16x4_f32(
                 false, a1, false, b, (short)0, c1, false, false);
    }

    const float bn    = bias[tileN + l16];
    const int   mhalf = (lane < 16) ? 0 : 8;
#pragma unroll
    for (int v = 0; v < 8; ++v) {
        out[(size_t)(tileM      + mhalf + v) * DIM + tileN + l16] = c0[v] + bn;
        out[(size_t)(tileM + 16 + mhalf + v) * DIM + tileN + l16] = c1[v] + bn;
    }
}

// ---------------------------------------------------------------------------
extern "C" void kernel_launch(void* const* d_in, const int* in_sizes, int n_in,
                              void* d_out, int out_size, void* d_ws, size_t ws_size,
                              hipStream_t stream) {
    const int*   user           = (const int*)  d_in[0];
    const int*   item           = (const int*)  d_in[1];
    const int*   behavior       = (const int*)  d_in[2];
    const int*   adj_rows       = (const int*)  d_in[3];
    const int*   adj_cols       = (const int*)  d_in[4];
    const float* adj_vals       = (const float*)d_in[5];
    const float* user_table     = (const float*)d_in[6];
    const float* item_table     = (const float*)d_in[7];
    const float* behavior_table = (const float*)d_in[8];
    const float* W              = (const float*)d_in[9];
    const float* b              = (const float*)d_in[10];
    float*       out            = (float*)d_out;

    // Workspace layout: x [N,D] then agg [N,D]  (2 * 51.2 MB)
    float* x   = (float*)d_ws;
    float* agg = x + (size_t)N_NODES * DIM;

    // 1) agg = 0                         (12.8M floats / 4 per thread)
    zero_f32_kernel<<<(N_NODES * DIM / 4 + 255) / 256, 256, 0, stream>>>(
        agg, N_NODES * DIM / 4);

    // 2) x = gather-sum of three tables  (100000 nodes * 32 lanes)
    embed_kernel<<<(N_NODES * 32 + 255) / 256, 256, 0, stream>>>(
        user, item, behavior, user_table, item_table, behavior_table, x);

    // 3) agg += scatter over 600K edges  (600000 edges * 32 lanes)
    spmm_scatter_kernel<<<(N_EDGES * 32 + 255) / 256, 256, 0, stream>>>(
        adj_rows, adj_cols, adj_vals, x, agg);

    // 4) out = agg @ W^T + b             (32-row panels, 8 waves/block)
    gemm_bias_wmma_kernel<<<N_NODES / ROWS_PER_BLOCK, 256, 0, stream>>>(
        agg, W, b, out);
}